// SetAbstraction_85873576116747
// MI455X (gfx1250) — compile-verified
//
#include <hip/hip_runtime.h>
#include <hip/hip_bf16.h>
#include <math.h>

// Problem constants (from reference setup_inputs)
#define BB 8
#define NN 8192
#define CF 64          // input feature channels
#define MM 2048        // FPS samples
#define KK 32          // neighbors
#define PTOT (BB * MM * KK)   // 524288 rows into the pointwise MLP

typedef __attribute__((ext_vector_type(2))) float v2f;
typedef __attribute__((ext_vector_type(8))) float v8f;

// ---------------------------------------------------------------------------
// 1) Farthest point sampling: one block per batch, points kept in registers.
// ---------------------------------------------------------------------------
__global__ __launch_bounds__(1024, 1)
void fps_kernel(const float* __restrict__ xyz, int* __restrict__ ci,
                float* __restrict__ newxyz) {
    int b = blockIdx.x;
    int t = threadIdx.x;                       // 0..1023, 8 points each
    const float* base = xyz + (size_t)b * NN * 3;

    float px[8], py[8], pz[8], dist[8];
#pragma unroll
    for (int i = 0; i < 8; i++) {
        int n = t * 8 + i;
        px[i] = base[n * 3 + 0];
        py[i] = base[n * 3 + 1];
        pz[i] = base[n * 3 + 2];
        dist[i] = 1e10f;
    }

    __shared__ float bc[3];
    __shared__ float wv[32];
    __shared__ int   wi[32];

    int far = 0;
    for (int s = 0; s < MM; s++) {
        // emit current 'far' (reference emits carry before update)
        if (t == (far >> 3)) {
            int j = far & 7;
            ci[b * MM + s] = far;
            float* o = newxyz + ((size_t)b * MM + s) * 3;
            o[0] = px[j]; o[1] = py[j]; o[2] = pz[j];
            bc[0] = px[j]; bc[1] = py[j]; bc[2] = pz[j];
        }
        __syncthreads();
        float cx = bc[0], cy = bc[1], cz = bc[2];

        float lv = -1.0f; int li = 0;
#pragma unroll
        for (int i = 0; i < 8; i++) {
            float dx = px[i] - cx, dy = py[i] - cy, dz = pz[i] - cz;
            float d = dx * dx + dy * dy + dz * dz;
            dist[i] = fminf(dist[i], d);
            if (dist[i] > lv) { lv = dist[i]; li = t * 8 + i; }
        }
        // wave32 argmax reduce (tie -> lower index, matches jnp.argmax)
#pragma unroll
        for (int off = 16; off > 0; off >>= 1) {
            float ov = __shfl_down(lv, off, 32);
            int   oi = __shfl_down(li, off, 32);
            if (ov > lv || (ov == lv && oi < li)) { lv = ov; li = oi; }
        }
        if ((t & 31) == 0) { wv[t >> 5] = lv; wi[t >> 5] = li; }
        __syncthreads();
        if (t < 32) {
            lv = wv[t]; li = wi[t];
#pragma unroll
            for (int off = 16; off > 0; off >>= 1) {
                float ov = __shfl_down(lv, off, 32);
                int   oi = __shfl_down(li, off, 32);
                if (ov > lv || (ov == lv && oi < li)) { lv = ov; li = oi; }
            }
            if (t == 0) { wi[0] = li; }
        }
        __syncthreads();
        far = wi[0];
    }
}

// ---------------------------------------------------------------------------
// 2) KNN: one block (256 threads) per query; d^2 in LDS, 32 selection rounds.
// ---------------------------------------------------------------------------
__global__ __launch_bounds__(256, 1)
void knn_kernel(const float* __restrict__ xyz, const float* __restrict__ newxyz,
                int* __restrict__ nn) {
    __shared__ float d2[NN];
    __shared__ float wv[8];
    __shared__ int   wi[8];

    int q = blockIdx.x;          // b*MM + m
    int b = q >> 11;             // / MM
    int t = threadIdx.x;
    const float* base = xyz + (size_t)b * NN * 3;
    float qx = newxyz[q * 3 + 0];
    float qy = newxyz[q * 3 + 1];
    float qz = newxyz[q * 3 + 2];
    float qq = qx * qx + qy * qy + qz * qz;

    for (int n = t; n < NN; n += 256) {
        float x = base[n * 3 + 0], y = base[n * 3 + 1], z = base[n * 3 + 2];
        float xx = x * x + y * y + z * z;
        d2[n] = qq + xx - 2.0f * (qx * x + qy * y + qz * z);
    }
    __syncthreads();

    for (int k = 0; k < KK; k++) {
        float lv = 3.4e38f; int li = 0x7fffffff;
        for (int n = t; n < NN; n += 256) {
            float v = d2[n];
            if (v < lv) { lv = v; li = n; }
        }
#pragma unroll
        for (int off = 16; off > 0; off >>= 1) {
            float ov = __shfl_down(lv, off, 32);
            int   oi = __shfl_down(li, off, 32);
            if (ov < lv || (ov == lv && oi < li)) { lv = ov; li = oi; }
        }
        if ((t & 31) == 0) { wv[t >> 5] = lv; wi[t >> 5] = li; }
        __syncthreads();
        if (t == 0) {
            float bv = wv[0]; int bi = wi[0];
#pragma unroll
            for (int w = 1; w < 8; w++) {
                if (wv[w] < bv || (wv[w] == bv && wi[w] < bi)) { bv = wv[w]; bi = wi[w]; }
            }
            nn[(size_t)q * KK + k] = bi;
            d2[bi] = 3.4e38f;
        }
        __syncthreads();
    }
}

// ---------------------------------------------------------------------------
// 3) WMMA GEMM:  Y[P x COUT] = X[P x CIN] @ W^T   (W is COUT x CIN, row-major)
//    One wave = one 16x16 output tile via V_WMMA_F32_16X16X4_F32.
//    GATHER: layer-0 A built on the fly (branchless: selects only, no exec
//    divergence around the WMMAs). CIN/COUT compile-time -> fully unrolled,
//    guard-free K loops. Per-channel sum/sum^2 accumulated for BatchNorm.
// ---------------------------------------------------------------------------
template <int CIN, int COUT, bool GATHER>
__global__ __launch_bounds__(256, 1)
void gemm_wmma(const float* __restrict__ A,
               const float* __restrict__ xyz, const float* __restrict__ feat,
               const float* __restrict__ nx, const int* __restrict__ nn,
               const float* __restrict__ W,
               float* __restrict__ Y, float* __restrict__ stats) {
    __shared__ float ls[256];
    int t = threadIdx.x;
    ls[t] = 0.0f;
    __syncthreads();

    constexpr int colTiles = COUT / 16;
    int waveId  = blockIdx.x * 8 + (t >> 5);
    int rowTile = waveId / colTiles;
    int colTile = waveId % colTiles;
    int lane  = t & 31;
    int lrow  = lane & 15;
    int khalf = lane >> 4;            // which K pair this lane supplies
    int row = rowTile * 16 + lrow;    // A row for this lane (all K steps)
    int col = colTile * 16 + lrow;    // B column for this lane

    const float* wrow = W + (size_t)col * CIN;
    v8f acc = {};

    if constexpr (GATHER) {
        // CIN == 67: channels = [dx,dy,dz, feat[0..63]], pad c=67 with 0.
        int pm  = row >> 5;           // / KK
        int b   = pm >> 11;           // / MM
        int nbr = nn[row];
        const float* p  = xyz + ((size_t)b * NN + nbr) * 3;
        const float* qa = nx + (size_t)pm * 3;
        float dx = p[0] - qa[0], dy = p[1] - qa[1], dz = p[2] - qa[2];
        const float* frow = feat + ((size_t)b * NN + nbr) * CF;

        {   // kb = 0: c = {0,1} (khalf=0) or {2,3} (khalf=1) -- pure selects
            v2f a, bm;
            a.x  = khalf ? dz : dx;
            a.y  = khalf ? frow[0] : dy;
            bm.x = wrow[khalf * 2];
            bm.y = wrow[khalf * 2 + 1];
            acc = __builtin_amdgcn_wmma_f32_16x16x4_f32(
                false, a, false, bm, (short)0, acc, false, false);
        }
#pragma unroll
        for (int kb = 4; kb < 64; kb += 4) {   // c in [4,63] -> plain loads
            int c0 = kb + khalf * 2;
            v2f a, bm;
            a.x  = frow[c0 - 3];
            a.y  = frow[c0 - 2];
            bm.x = wrow[c0];
            bm.y = wrow[c0 + 1];
            acc = __builtin_amdgcn_wmma_f32_16x16x4_f32(
                false, a, false, bm, (short)0, acc, false, false);
        }
        {   // kb = 64 tail: c = {64,65} or {66, 67->pad 0}
            v2f a, bm;
            a.x  = frow[khalf ? 63 : 61];
            a.y  = khalf ? 0.0f : frow[62];
            bm.x = wrow[khalf ? 66 : 64];
            bm.y = khalf ? 0.0f : wrow[65];
            acc = __builtin_amdgcn_wmma_f32_16x16x4_f32(
                false, a, false, bm, (short)0, acc, false, false);
        }
    } else {
        // CIN multiple of 4; rows 8B-aligned -> vector b64 loads, no guards.
        const float* arow = A + (size_t)row * CIN;
#pragma unroll
        for (int kb = 0; kb < CIN; kb += 4) {
            int c0 = kb + khalf * 2;
            v2f a  = *(const v2f*)(arow + c0);
            v2f bm = *(const v2f*)(wrow + c0);
            acc = __builtin_amdgcn_wmma_f32_16x16x4_f32(
                false, a, false, bm, (short)0, acc, false, false);
        }
    }

    // store 16x16 tile + accumulate BN statistics
    float s = 0.0f, sq = 0.0f;
#pragma unroll
    for (int v = 0; v < 8; v++) {
        int m = rowTile * 16 + v + khalf * 8;   // C/D layout: lanes 16-31 hold M+8
        float val = acc[v];
        Y[(size_t)m * COUT + col] = val;
        s += val; sq += val * val;
    }
    s  += __shfl_down(s, 16, 32);
    sq += __shfl_down(sq, 16, 32);
    if (lane < 16) {
        atomicAdd(&ls[col], s);
        atomicAdd(&ls[128 + col], sq);
    }
    __syncthreads();
    if (t < COUT) {
        atomicAdd(&stats[t], ls[t]);
        atomicAdd(&stats[128 + t], ls[128 + t]);
    }
}

// ---------------------------------------------------------------------------
// 4) Training-mode BatchNorm + exact GELU, in place.
// ---------------------------------------------------------------------------
__global__ void bn_gelu(float* __restrict__ H, const float* __restrict__ stats,
                        const float* __restrict__ g, const float* __restrict__ be,
                        int C, int total) {
    int i = blockIdx.x * 256 + threadIdx.x;
    if (i >= total) return;
    int c = i & (C - 1);                       // C is a power of two (64/128)
    const float invP = 1.0f / (float)PTOT;
    float mean = stats[c] * invP;
    float var  = stats[128 + c] * invP - mean * mean;   // biased variance
    float x = (H[i] - mean) * rsqrtf(var + 1e-5f) * g[c] + be[c];
    H[i] = 0.5f * x * (1.0f + erff(x * 0.70710678118654752f));
}

// ---------------------------------------------------------------------------
// 5) Max over the k=32 neighbors -> pooled output.
// ---------------------------------------------------------------------------
__global__ void maxk_kernel(const float* __restrict__ Y2, float* __restrict__ out) {
    int i = blockIdx.x * 256 + threadIdx.x;     // over B*M*128
    if (i >= BB * MM * 128) return;
    int pm = i >> 7;
    int c  = i & 127;
    const float* p = Y2 + ((size_t)pm * KK) * 128 + c;
    float best = -3.4e38f;
#pragma unroll
    for (int k = 0; k < KK; k++) best = fmaxf(best, p[(size_t)k * 128]);
    out[i] = best;
}

// ---------------------------------------------------------------------------
extern "C" void kernel_launch(void* const* d_in, const int* in_sizes, int n_in,
                              void* d_out, int out_size, void* d_ws, size_t ws_size,
                              hipStream_t stream) {
    const float* xyz  = (const float*)d_in[0];
    const float* feat = (const float*)d_in[1];
    const float* W0 = (const float*)d_in[2];
    const float* g0 = (const float*)d_in[3];
    const float* b0 = (const float*)d_in[4];
    const float* W1 = (const float*)d_in[5];
    const float* g1 = (const float*)d_in[6];
    const float* b1 = (const float*)d_in[7];
    const float* W2 = (const float*)d_in[8];
    const float* g2 = (const float*)d_in[9];
    const float* b2 = (const float*)d_in[10];

    float* out     = (float*)d_out;
    float* newxyz  = out;                       // (B, M, 3)
    float* pooled  = out + (size_t)BB * MM * 3; // (B, M, 128)

    // workspace carve-out
    char* w = (char*)d_ws;
    int*   ci    = (int*)w;   w += (size_t)BB * MM * sizeof(int);
    int*   nn    = (int*)w;   w += (size_t)PTOT * sizeof(int);
    float* stats = (float*)w; w += 256 * sizeof(float);
    float* Y0    = (float*)w; w += (size_t)PTOT * 64  * sizeof(float);
    float* Y1    = (float*)w; w += (size_t)PTOT * 64  * sizeof(float);
    float* Y2    = (float*)w; w += (size_t)PTOT * 128 * sizeof(float);

    fps_kernel<<<BB, 1024, 0, stream>>>(xyz, ci, newxyz);
    knn_kernel<<<BB * MM, 256, 0, stream>>>(xyz, newxyz, nn);

    const int rowTiles = PTOT / 16;             // 32768

    // Layer 0: gathered input (Cin=67) -> 64 channels
    hipMemsetAsync(stats, 0, 256 * sizeof(float), stream);
    gemm_wmma<67, 64, true><<<(rowTiles * (64 / 16)) / 8, 256, 0, stream>>>(
        nullptr, xyz, feat, newxyz, nn, W0, Y0, stats);
    bn_gelu<<<(PTOT * 64 + 255) / 256, 256, 0, stream>>>(Y0, stats, g0, b0, 64, PTOT * 64);

    // Layer 1: 64 -> 64
    hipMemsetAsync(stats, 0, 256 * sizeof(float), stream);
    gemm_wmma<64, 64, false><<<(rowTiles * (64 / 16)) / 8, 256, 0, stream>>>(
        Y0, nullptr, nullptr, nullptr, nullptr, W1, Y1, stats);
    bn_gelu<<<(PTOT * 64 + 255) / 256, 256, 0, stream>>>(Y1, stats, g1, b1, 64, PTOT * 64);

    // Layer 2: 64 -> 128
    hipMemsetAsync(stats, 0, 256 * sizeof(float), stream);
    gemm_wmma<64, 128, false><<<(rowTiles * (128 / 16)) / 8, 256, 0, stream>>>(
        Y1, nullptr, nullptr, nullptr, nullptr, W2, Y2, stats);
    bn_gelu<<<(PTOT * 128 + 255) / 256, 256, 0, stream>>>(Y2, stats, g2, b2, 128, PTOT * 128);

    maxk_kernel<<<(BB * MM * 128 + 255) / 256, 256, 0, stream>>>(Y2, pooled);
}